// TriangleScene_4217657884862
// MI455X (gfx1250) — compile-verified
//
#include <hip/hip_runtime.h>
#include <math.h>

// 10 * FLT_EPSILON — shared by EPS, HIT_TOL, MIN_LEN in the reference
#define EPS_F 1.1920928955078125e-06f
#define ORDER 2
#define NT_PER_WAVE 10  // triangle 16-tiles processed per wave (A stays resident)

typedef float v2f __attribute__((ext_vector_type(2)));
typedef float v8f __attribute__((ext_vector_type(8)));

struct F3 { float x, y, z; };
__device__ __forceinline__ F3 sub3(F3 a, F3 b) { return {a.x - b.x, a.y - b.y, a.z - b.z}; }
__device__ __forceinline__ F3 add3(F3 a, F3 b) { return {a.x + b.x, a.y + b.y, a.z + b.z}; }
__device__ __forceinline__ F3 scl3(float s, F3 a) { return {s * a.x, s * a.y, s * a.z}; }
__device__ __forceinline__ float dot3(F3 a, F3 b) { return a.x * b.x + a.y * b.y + a.z * b.z; }
__device__ __forceinline__ F3 cross3(F3 a, F3 b) {
    return {a.y * b.z - a.z * b.y, a.z * b.x - a.x * b.z, a.x * b.y - a.y * b.x};
}
__device__ __forceinline__ F3 loadv3(const float* __restrict__ p, int i) {
    return {p[3 * i + 0], p[3 * i + 1], p[3 * i + 2]};
}

// Reference Moller-Trumbore (used only for the 2 per-path "inside" tests)
__device__ __forceinline__ bool mt_hit(F3 o, F3 d, F3 A, F3 B, F3 Cv) {
    F3 e1 = sub3(B, A), e2 = sub3(Cv, A);
    F3 h = cross3(d, e2);
    float a = dot3(e1, h);
    bool cond = fabsf(a) > EPS_F;
    float f = 1.0f / (cond ? a : 1.0f);
    F3 s = sub3(o, A);
    float u = f * dot3(s, h);
    F3 q = cross3(s, e1);
    float v = f * dot3(d, q);
    float t = f * dot3(e2, q);
    // bitwise combine: keep it branchless
    return (bool)((int)cond & (int)(u >= 0.0f) & (int)(u <= 1.0f) & (int)(v >= 0.0f) &
                  (int)(u + v <= 1.0f) & (int)(t > EPS_F));
}

// ---------------------------------------------------------------------------
// Kernel A: per-path image-method construction + per-path predicates.
// ---------------------------------------------------------------------------
__global__ void paths_kernel(const float* __restrict__ mv, const float* __restrict__ mn,
                             const float* __restrict__ fromv, const float* __restrict__ tov,
                             const int* __restrict__ tris, const int* __restrict__ cands,
                             int tx, int rx, int C,
                             float* __restrict__ out_full, float* __restrict__ out_obj,
                             float* __restrict__ rayMat, int* __restrict__ preMask,
                             int* __restrict__ flags) {
    int P = tx * rx * C;
    int p = blockIdx.x * blockDim.x + threadIdx.x;
    if (p >= P) return;
    int c = p % C;
    int rI = (p / C) % rx;
    int tI = p / (C * rx);

    int cand[ORDER];
    F3 vA[ORDER], vB[ORDER], vC[ORDER], nrm[ORDER];
#pragma unroll
    for (int k = 0; k < ORDER; ++k) {
        cand[k] = cands[c * ORDER + k];
        int i0 = tris[cand[k] * 3 + 0];
        int i1 = tris[cand[k] * 3 + 1];
        int i2 = tris[cand[k] * 3 + 2];
        vA[k] = loadv3(mv, i0);
        vB[k] = loadv3(mv, i1);
        vC[k] = loadv3(mv, i2);
        nrm[k] = loadv3(mn, cand[k]);
    }
    F3 f = loadv3(fromv, tI);
    F3 t = loadv3(tov, rI);

    // forward scan: mirror images of the source
    F3 img[ORDER];
    F3 carry = f;
#pragma unroll
    for (int k = 0; k < ORDER; ++k) {
        carry = sub3(carry, scl3(2.0f * dot3(sub3(carry, vA[k]), nrm[k]), nrm[k]));
        img[k] = carry;
    }
    // backward scan: intersect with mirror planes
    F3 pts[ORDER + 2];
    pts[0] = f;
    pts[ORDER + 1] = t;
    F3 cb = t;
#pragma unroll
    for (int k = ORDER - 1; k >= 0; --k) {
        F3 u = sub3(cb, img[k]);
        float tp = dot3(sub3(vA[k], img[k]), nrm[k]) / dot3(u, nrm[k]);
        cb = add3(img[k], scl3(tp, u));
        pts[k + 1] = cb;
    }

    // inside_triangles: segment k vs candidate triangle k
    bool inside = true;
#pragma unroll
    for (int k = 0; k < ORDER; ++k) {
        F3 d = sub3(pts[k + 1], pts[k]);
        inside = inside && mt_hit(pts[k], d, vA[k], vB[k], vC[k]);
    }
    // valid_reflections: same-side test
    bool valid = true;
#pragma unroll
    for (int k = 0; k < ORDER; ++k) {
        float dp = dot3(sub3(pts[k], vA[k]), nrm[k]);
        float dn = dot3(sub3(pts[k + 2], vA[k]), nrm[k]);
        valid = valid && (dp * dn >= 0.0f);
    }
    // too_small: any degenerate segment
    bool small = false;
#pragma unroll
    for (int s = 0; s < ORDER + 1; ++s) {
        F3 d = sub3(pts[s + 1], pts[s]);
        small = small || (dot3(d, d) < EPS_F);
    }

    preMask[p] = (inside && valid && !small) ? 1 : 0;
    flags[p] = 0;

    // outputs: full (p*12 floats) and objects (p*4 floats)
#pragma unroll
    for (int i = 0; i < ORDER + 2; ++i) {
        out_full[p * 12 + i * 3 + 0] = pts[i].x;
        out_full[p * 12 + i * 3 + 1] = pts[i].y;
        out_full[p * 12 + i * 3 + 2] = pts[i].z;
    }
    out_obj[p * 4 + 0] = (float)tI;
#pragma unroll
    for (int k = 0; k < ORDER; ++k) out_obj[p * 4 + 1 + k] = (float)cand[k];
    out_obj[p * 4 + ORDER + 1] = (float)rI;

    // ray matrix rows for the WMMA GEMM: row 2r = (o,1), row 2r+1 = (d,0)
#pragma unroll
    for (int s = 0; s < ORDER + 1; ++s) {
        int r = p * (ORDER + 1) + s;
        F3 o = pts[s];
        F3 d = sub3(pts[s + 1], pts[s]);
        float* row = rayMat + (size_t)r * 8;
        row[0] = o.x; row[1] = o.y; row[2] = o.z; row[3] = 1.0f;
        row[4] = d.x; row[5] = d.y; row[6] = d.z; row[7] = 0.0f;
    }
}

// ---------------------------------------------------------------------------
// Kernel B: per-triangle Wald precompute — three homogeneous 4-vectors.
// ---------------------------------------------------------------------------
__global__ void tri_kernel(const float* __restrict__ mv, const int* __restrict__ tris, int T,
                           float* __restrict__ tbn, float* __restrict__ tbu,
                           float* __restrict__ tbv) {
    int j = blockIdx.x * blockDim.x + threadIdx.x;
    if (j >= T) return;
    F3 A = loadv3(mv, tris[j * 3 + 0]);
    F3 B = loadv3(mv, tris[j * 3 + 1]);
    F3 Cv = loadv3(mv, tris[j * 3 + 2]);
    F3 e1 = sub3(B, A), e2 = sub3(Cv, A);
    F3 n = cross3(e1, e2);
    float nn = dot3(n, n);
    float inv = (nn > 0.0f) ? (1.0f / nn) : 0.0f;
    F3 m1 = scl3(inv, cross3(e2, n));   // u = m1 . (P - v0)
    F3 m2 = scl3(inv, cross3(n, e1));   // v = m2 . (P - v0)
    tbn[j * 4 + 0] = n.x;  tbn[j * 4 + 1] = n.y;  tbn[j * 4 + 2] = n.z;  tbn[j * 4 + 3] = -dot3(n, A);
    tbu[j * 4 + 0] = m1.x; tbu[j * 4 + 1] = m1.y; tbu[j * 4 + 2] = m1.z; tbu[j * 4 + 3] = -dot3(m1, A);
    tbv[j * 4 + 0] = m2.x; tbv[j * 4 + 1] = m2.y; tbv[j * 4 + 2] = m2.z; tbv[j * 4 + 3] = -dot3(m2, A);
}

// ---------------------------------------------------------------------------
// Kernel C: blocked test as K=4 fp32 WMMA GEMMs.
// One wave = 8 rays (16 interleaved O/D rows), A fragment resident, looping
// over NT_PER_WAVE triangle 16-tiles (3 WMMAs each). Fully branchless
// epilogue; per-lane 4-bit blocked mask; wave-level ballot reduction with
// <=8 atomics per wave.
// ---------------------------------------------------------------------------
__global__ void __launch_bounds__(256) blocked_kernel(
    const float* __restrict__ rayMat, const float* __restrict__ tbn,
    const float* __restrict__ tbu, const float* __restrict__ tbv,
    int R, int T, int ntiles, int ntGroups, int totalWaves, int* __restrict__ flags) {
    int wave = blockIdx.x * (blockDim.x >> 5) + (threadIdx.x >> 5);
    if (wave >= totalWaves) return;  // wave-uniform: EXEC stays all-ones for WMMA
    int lane = threadIdx.x & 31;
    int mt = wave / ntGroups;
    int ng = wave % ntGroups;
    int half = lane >> 4;
    int lm = lane & 15;
    int k0 = half * 2;

    // A fragment (f32 16x4 layout): lanes 0-15 row M with K0/K1, lanes 16-31 K2/K3
    int rows = 2 * R;
    int row = mt * 16 + lm;
    if (row >= rows) row = rows - 1;
    const float* rrow = rayMat + (size_t)row * 4;
    v2f a;
    a.x = rrow[k0];
    a.y = rrow[k0 + 1];

    int bmask = 0;  // bit i: ray (mt*8 + half*4 + i) blocked by any triangle seen
    int ntBegin = ng * NT_PER_WAVE;
    int ntEnd = ntBegin + NT_PER_WAVE;
    if (ntEnd > ntiles) ntEnd = ntiles;

    for (int nt = ntBegin; nt < ntEnd; ++nt) {
        int j = nt * 16 + lm;
        int jc = (j < T) ? j : (T - 1);
        __builtin_prefetch(tbn + (size_t)jc * 4, 0, 0);
        v2f bn, bu, bv;
        bn.x = tbn[jc * 4 + k0]; bn.y = tbn[jc * 4 + k0 + 1];
        bu.x = tbu[jc * 4 + k0]; bu.y = tbu[jc * 4 + k0 + 1];
        bv.x = tbv[jc * 4 + k0]; bv.y = tbv[jc * 4 + k0 + 1];

        v8f zero = {};
        // (neg_a, A, neg_b, B, c_mod, C, reuse_a, reuse_b)
        v8f dn = __builtin_amdgcn_wmma_f32_16x16x4_f32(false, a, false, bn, (short)0, zero, false, false);
        v8f du = __builtin_amdgcn_wmma_f32_16x16x4_f32(false, a, false, bu, (short)0, zero, false, false);
        v8f dv = __builtin_amdgcn_wmma_f32_16x16x4_f32(false, a, false, bv, (short)0, zero, false, false);

        int jvalid = (int)(j < T);
#pragma unroll
        for (int i = 0; i < 4; ++i) {
            float pon = dn[2 * i], pdn = dn[2 * i + 1];  // n.o - n.v0 ,  n.d
            float pou = du[2 * i], pdu = du[2 * i + 1];
            float pov = dv[2 * i], pdv = dv[2 * i + 1];
            int cond = (int)(fabsf(pdn) > EPS_F);
            // fast reciprocal (v_rcp_f32): ~1 ulp, plenty for tolerance predicates
            float t = -pon * __builtin_amdgcn_rcpf(cond ? pdn : 1.0f);
            float u = fmaf(t, pdu, pou);
            float v = fmaf(t, pdv, pov);
            // bitwise & (no short-circuit) -> straight v_cmp/s_and chain, no exec branches
            int blocked = jvalid & cond & (int)(u >= 0.0f) & (int)(u <= 1.0f) &
                          (int)(v >= 0.0f) & (int)(u + v <= 1.0f) & (int)(t > EPS_F) &
                          (int)(t < 1.0f - EPS_F);
            bmask |= blocked << i;
        }
    }

    // Wave-level OR per ray: lanes 0-15 share rays mt*8+i, lanes 16-31 rays mt*8+4+i
#pragma unroll
    for (int i = 0; i < 4; ++i) {
        unsigned m = __builtin_amdgcn_ballot_w32((bmask >> i) & 1);
        if (lane == 0 && (m & 0xFFFFu)) {
            int ray = mt * 8 + i;
            if (ray < R) atomicOr(&flags[ray / (ORDER + 1)], 1);
        }
        if (lane == 16 && (m >> 16)) {
            int ray = mt * 8 + 4 + i;
            if (ray < R) atomicOr(&flags[ray / (ORDER + 1)], 1);
        }
    }
}

// ---------------------------------------------------------------------------
// Kernel D: final mask = preMask && !blocked
// ---------------------------------------------------------------------------
__global__ void finalize_kernel(const int* __restrict__ preMask, const int* __restrict__ flags,
                                float* __restrict__ out_mask, int P) {
    int p = blockIdx.x * blockDim.x + threadIdx.x;
    if (p >= P) return;
    out_mask[p] = (preMask[p] && !flags[p]) ? 1.0f : 0.0f;
}

extern "C" void kernel_launch(void* const* d_in, const int* in_sizes, int n_in, void* d_out,
                              int out_size, void* d_ws, size_t ws_size, hipStream_t stream) {
    const float* mesh_vertices = (const float*)d_in[0];
    const float* mesh_normals = (const float*)d_in[1];
    const float* from_vertices = (const float*)d_in[2];
    const float* to_vertices = (const float*)d_in[3];
    const int* mesh_triangles = (const int*)d_in[4];
    const int* path_candidates = (const int*)d_in[5];

    const int tx = in_sizes[2] / 3;
    const int rx = in_sizes[3] / 3;
    const int T = in_sizes[4] / 3;
    const int C = in_sizes[5] / ORDER;
    const int P = tx * rx * C;          // paths
    const int R = P * (ORDER + 1);      // ray segments
    const int rows = 2 * R;             // interleaved O/D rows for the GEMM

    // workspace carve-up
    char* ws = (char*)d_ws;
    size_t off = 0;
    float* rayMat = (float*)(ws + off); off += (size_t)rows * 4 * sizeof(float);
    float* tbn = (float*)(ws + off);    off += (size_t)T * 4 * sizeof(float);
    float* tbu = (float*)(ws + off);    off += (size_t)T * 4 * sizeof(float);
    float* tbv = (float*)(ws + off);    off += (size_t)T * 4 * sizeof(float);
    int* flags = (int*)(ws + off);      off += (size_t)P * sizeof(int);
    int* preMask = (int*)(ws + off);    off += (size_t)P * sizeof(int);
    (void)ws_size;

    // output carve-up (floats): full | mask | objects
    float* out = (float*)d_out;
    float* out_full = out;
    float* out_mask = out + (size_t)P * (ORDER + 2) * 3;
    float* out_obj = out_mask + (size_t)P;
    (void)out_size;

    paths_kernel<<<(P + 255) / 256, 256, 0, stream>>>(
        mesh_vertices, mesh_normals, from_vertices, to_vertices, mesh_triangles, path_candidates,
        tx, rx, C, out_full, out_obj, rayMat, preMask, flags);

    tri_kernel<<<(T + 255) / 256, 256, 0, stream>>>(mesh_vertices, mesh_triangles, T, tbn, tbu, tbv);

    const int mtiles = (rows + 15) / 16;
    const int ntiles = (T + 15) / 16;
    const int ntGroups = (ntiles + NT_PER_WAVE - 1) / NT_PER_WAVE;
    const int totalWaves = mtiles * ntGroups;
    const int wavesPerBlock = 256 / 32;
    blocked_kernel<<<(totalWaves + wavesPerBlock - 1) / wavesPerBlock, 256, 0, stream>>>(
        rayMat, tbn, tbu, tbv, R, T, ntiles, ntGroups, totalWaves, flags);

    finalize_kernel<<<(P + 255) / 256, 256, 0, stream>>>(preMask, flags, out_mask, P);
}